// LaplaceNeuralOperator_15951508537782
// MI455X (gfx1250) — compile-verified
//
#include <hip/hip_runtime.h>
#include <hip/hip_bf16.h>
#include <math.h>

typedef __attribute__((ext_vector_type(2))) float v2f;
typedef __attribute__((ext_vector_type(8))) float v8f;

#define Bsz 8
#define Hd  64
#define Pd  16
#define Nd  1024

__device__ __forceinline__ float fast_rcp(float x) {
#if __has_builtin(__builtin_amdgcn_rcpf)
  return __builtin_amdgcn_rcpf(x);
#else
  return 1.0f / x;
#endif
}

__device__ __forceinline__ float gelu_f(float x) {
  const float k0 = 0.7978845608028654f; // sqrt(2/pi)
  float x3 = x * x * x;
  return 0.5f * x * (1.0f + tanhf(k0 * (x + 0.044715f * x3)));
}

// ---------------------------------------------------------------------------
// K0: transpose pole/residue params (i,o,p) -> (p,i,o) for coalesced access,
//     and zero the out_res2 accumulator (re-zeroed every launch: graph-safe).
// ---------------------------------------------------------------------------
__global__ __launch_bounds__(256) void prep_params_kernel(
    const float* __restrict__ wp, const float* __restrict__ wpi,
    const float* __restrict__ wr, const float* __restrict__ wri,
    float* __restrict__ pre_t, float* __restrict__ pim_t,
    float* __restrict__ rre_t, float* __restrict__ rim_t,
    float* __restrict__ a2re, float* __restrict__ a2im)
{
  int t = blockIdx.x * 256 + threadIdx.x;      // 65536 = 16*64*64
  int o = t & 63;
  int i = (t >> 6) & 63;
  int p = t >> 12;
  int src = (i * Hd + o) * Pd + p;
  pre_t[t] = wp[src];
  pim_t[t] = wpi[src];
  rre_t[t] = wr[src];
  rim_t[t] = wri[src];
  if (t < Bsz * Hd * Pd) { a2re[t] = 0.0f; a2im[t] = 0.0f; }
}

// ---------------------------------------------------------------------------
// K1: lift  h[b,i,n] = sum_c x[b,c,n]*W0[c,i] + b0[i]   (CIN=2)
// ---------------------------------------------------------------------------
__global__ __launch_bounds__(256) void lift_kernel(
    const float* __restrict__ x, const float* __restrict__ W0,
    const float* __restrict__ b0, float* __restrict__ h)
{
  int g = blockIdx.x * 256 + threadIdx.x;      // 524288 = 8*64*1024
  int n = g & 1023;
  int bi = g >> 10;
  int i = bi & 63;
  int b = bi >> 6;
  float v = x[(b * 2 + 0) * Nd + n] * W0[0 * Hd + i]
          + x[(b * 2 + 1) * Nd + n] * W0[1 * Hd + i]
          + b0[i];
  h[g] = v;
}

// ---------------------------------------------------------------------------
// K2/K4: 1024-point Stockham radix-2 FFT in LDS (one row per block).
// sign=-1,scale=1 : forward FFT.  sign=+1,scale=1/N : inverse FFT.
// in_im==nullptr -> real input.   out_im may be discarded by caller.
// ---------------------------------------------------------------------------
__global__ __launch_bounds__(256) void fft1024_kernel(
    const float* __restrict__ in_re, const float* __restrict__ in_im,
    float* __restrict__ out_re, float* __restrict__ out_im,
    float sign, float scale)
{
  __shared__ float2 bufA[1024];
  __shared__ float2 bufB[1024];
  const int row = blockIdx.x;
  const int tid = threadIdx.x;

  for (int j = tid; j < 1024; j += 256) {
    float re = in_re[row * 1024 + j];
    float im = in_im ? in_im[row * 1024 + j] : 0.0f;
    bufA[j] = make_float2(re, im);
  }
  __syncthreads();

  float2* src = bufA;
  float2* dst = bufB;
  int m = 1;
  for (int s = 0; s < 10; ++s) {
    for (int u = tid; u < 512; u += 256) {
      int j = u >> s;
      int k = u & (m - 1);
      int jm = j * m;
      float2 c0 = src[jm + k];
      float2 c1 = src[jm + k + 512];
      float ang = sign * 6.283185307179586f * (float)jm * (1.0f / 1024.0f);
      float sw, cw;
      __sincosf(ang, &sw, &cw);
      float dx = c0.x - c1.x;
      float dy = c0.y - c1.y;
      dst[2 * jm + k]     = make_float2(c0.x + c1.x, c0.y + c1.y);
      dst[2 * jm + k + m] = make_float2(cw * dx - sw * dy, cw * dy + sw * dx);
    }
    __syncthreads();
    float2* t = src; src = dst; dst = t;
    m <<= 1;
  }

  for (int j = tid; j < 1024; j += 256) {
    out_re[row * 1024 + j] = src[j].x * scale;
    if (out_im) out_im[row * 1024 + j] = src[j].y * scale;
  }
}

// ---------------------------------------------------------------------------
// K3: per-frequency complex Cauchy mixing with f32 WMMA.
// A (16x64): rows 0-7 = Re(alpha[b]), rows 8-15 = Im(alpha[b]) at frequency n.
// B (64x64): M_p[i][o] = res[i,o,p] / (lam_n - pole[i,o,p])  (Re and Im).
// Two wmma chains give all 4 real products; combined via shfl_xor(16).
// out_res1 accumulated over p in regs; out_res2 reduced over n via f32 atomics.
// Block = 128 thr (4 waves, one 16-wide o-tile each), handles 4 frequencies.
// ---------------------------------------------------------------------------
__global__ __launch_bounds__(128) void laplace_freq_kernel(
    const float* __restrict__ alpha_re, const float* __restrict__ alpha_im,
    const float* __restrict__ pre_t, const float* __restrict__ pim_t,
    const float* __restrict__ rre_t, const float* __restrict__ rim_t,
    float* __restrict__ r1_re, float* __restrict__ r1_im,
    float* __restrict__ a2_re, float* __restrict__ a2_im)
{
  __shared__ float ldsA[4 * 64 * 16];   // [q][i(K)][m(M)]
  const int tid  = threadIdx.x;
  const int wave = tid >> 5;            // o-tile
  const int lane = tid & 31;
  const int c    = lane & 15;
  const int hi   = lane >> 4;           // 0: lanes 0-15, 1: lanes 16-31
  const int n0   = blockIdx.x * 4;
  const int o    = wave * 16 + c;

  // stage packed alpha tiles for the 4 frequencies
  for (int idx = tid; idx < 4096; idx += 128) {
    int q = idx >> 10, rem = idx & 1023, i = rem >> 4, m = rem & 15;
    int n = n0 + q;
    const float* sp = (m < 8) ? &alpha_re[(m * Hd + i) * Nd + n]
                              : &alpha_im[((m - 8) * Hd + i) * Nd + n];
    ldsA[idx] = *sp;
  }
  __syncthreads();

  float lamIm[4];
#pragma unroll
  for (int q = 0; q < 4; ++q) {
    int n = n0 + q;
    int f = (n < 512) ? n : (n - 1024);       // fftfreq(N,1/N) integer freqs
    lamIm[q] = 6.283185307179586f * (float)f;
  }

  v8f rAcc[4] = {};

  for (int p = 0; p < Pd; ++p) {
    v8f acc0[4] = {};   // A x Re(Mp)
    v8f acc1[4] = {};   // A x Im(Mp)

    for (int kk = 0; kk < 16; ++kk) {
      int i0 = 4 * kk + 2 * hi;                 // K slots for this lane half
      int base0 = (p * Hd + i0) * Hd + o;
      int base1 = base0 + Hd;                   // i0+1 row
      float pr0 = pre_t[base0], pi0 = pim_t[base0];
      float rr0 = rre_t[base0], ri0 = rim_t[base0];
      float pr1 = pre_t[base1], pi1 = pim_t[base1];
      float rr1 = rre_t[base1], ri1 = rim_t[base1];
      // q-invariant precompute (lam only changes the imaginary denominator)
      float dre0 = -pr0, dre1 = -pr1;
      float dre0sq = dre0 * dre0, dre1sq = dre1 * dre1;
      float nre0c = rr0 * dre0, nim0c = ri0 * dre0;
      float nre1c = rr1 * dre1, nim1c = ri1 * dre1;

#pragma unroll
      for (int q = 0; q < 4; ++q) {
        float dim0 = lamIm[q] - pi0;
        float dim1 = lamIm[q] - pi1;
        float inv0 = fast_rcp(dre0sq + dim0 * dim0);
        float inv1 = fast_rcp(dre1sq + dim1 * dim1);
        v2f bre, bim;
        bre.x = (nre0c + ri0 * dim0) * inv0;    // Re(res/(lam-pole))
        bim.x = (nim0c - rr0 * dim0) * inv0;    // Im(res/(lam-pole))
        bre.y = (nre1c + ri1 * dim1) * inv1;
        bim.y = (nim1c - rr1 * dim1) * inv1;
        v2f a;
        a.x = ldsA[q * 1024 + i0 * 16 + c];
        a.y = ldsA[q * 1024 + (i0 + 1) * 16 + c];
        acc0[q] = __builtin_amdgcn_wmma_f32_16x16x4_f32(
            false, a, false, bre, (short)0, acc0[q], false, false);
        acc1[q] = __builtin_amdgcn_wmma_f32_16x16x4_f32(
            false, a, false, bim, (short)0, acc1[q], false, false);
      }
    }

    // combine: lanes<16 get Re(S)=ReRe-ImIm, lanes>=16 get Im(S)=ImRe+ReIm
    float r2[8];
#pragma unroll
    for (int e = 0; e < 8; ++e) r2[e] = 0.0f;

#pragma unroll
    for (int q = 0; q < 4; ++q) {
#pragma unroll
      for (int e = 0; e < 8; ++e) {
        float s1 = __shfl_xor(acc1[q][e], 16, 32);
        float v  = hi ? (acc0[q][e] + s1) : (acc0[q][e] - s1);
        rAcc[q][e] += v;
        r2[e] += v;
      }
    }

    float* a2p = hi ? a2_im : a2_re;
#pragma unroll
    for (int e = 0; e < 8; ++e)
      atomicAdd(&a2p[(e * Hd + o) * Pd + p], -r2[e]);   // out_res2 = -sum_n S
  }

  float* r1p = hi ? r1_im : r1_re;
#pragma unroll
  for (int q = 0; q < 4; ++q)
#pragma unroll
    for (int e = 0; e < 8; ++e)
      r1p[(e * Hd + o) * Nd + (n0 + q)] = rAcc[q][e];   // out_res1 = sum_p S
}

// ---------------------------------------------------------------------------
// K5: E[o,p,n] = sum_i exp(pole[i,o,p] * t[n]),  t[n] = n/(N-1)
// ---------------------------------------------------------------------------
__global__ __launch_bounds__(256) void eterm_kernel(
    const float* __restrict__ pre_t, const float* __restrict__ pim_t,
    float* __restrict__ E_re, float* __restrict__ E_im)
{
  int g = blockIdx.x * 256 + threadIdx.x;   // 1048576 = 64*16*1024
  int n = g & 1023;
  int op = g >> 10;
  int p = op & 15;
  int o = op >> 4;
  float t = (float)n * (1.0f / 1023.0f);
  float sre = 0.0f, sim = 0.0f;
  for (int i = 0; i < 64; ++i) {
    int idx = (p * Hd + i) * Hd + o;
    float prr = pre_t[idx];
    float pii = pim_t[idx];
    float mag = __expf(prr * t);
    float s, cc;
    __sincosf(pii * t, &s, &cc);
    sre += mag * cc;
    sim += mag * s;
  }
  E_re[g] = sre;
  E_im[g] = sim;                            // layout (o*16+p)*1024+n == g
}

// ---------------------------------------------------------------------------
// K6: h2 = gelu( x1 + Re(out_res2 * E)/N + h@Wl + bl )
// ---------------------------------------------------------------------------
__global__ __launch_bounds__(256) void combine_kernel(
    const float* __restrict__ x1_re,
    const float* __restrict__ a2re, const float* __restrict__ a2im,
    const float* __restrict__ E_re, const float* __restrict__ E_im,
    const float* __restrict__ h, const float* __restrict__ Wl,
    const float* __restrict__ bl, float* __restrict__ h2)
{
  int g = blockIdx.x * 256 + threadIdx.x;   // 524288
  int n = g & 1023;
  int bo = g >> 10;
  int o = bo & 63;
  int b = bo >> 6;

  float x2 = 0.0f;
#pragma unroll
  for (int p = 0; p < Pd; ++p) {
    float r2r = a2re[(b * Hd + o) * Pd + p];
    float r2i = a2im[(b * Hd + o) * Pd + p];
    float er  = E_re[(o * Pd + p) * Nd + n];
    float ei  = E_im[(o * Pd + p) * Nd + n];
    x2 += r2r * er - r2i * ei;
  }
  x2 *= (1.0f / 1024.0f);

  float skip = bl[o];
  for (int i = 0; i < 64; ++i)
    skip += h[(b * Hd + i) * Nd + n] * Wl[i * Hd + o];

  h2[g] = gelu_f(x1_re[g] + x2 + skip);
}

// ---------------------------------------------------------------------------
// K7: out[b,0,n] = gelu(h2@W1 + b1) @ W2 + b2
// ---------------------------------------------------------------------------
__global__ __launch_bounds__(256) void project_kernel(
    const float* __restrict__ h2, const float* __restrict__ W1,
    const float* __restrict__ b1, const float* __restrict__ W2,
    const float* __restrict__ b2, float* __restrict__ out)
{
  int g = blockIdx.x * 256 + threadIdx.x;   // 8192 = 8*1024
  int n = g & 1023;
  int b = g >> 10;
  float hv[64];
#pragma unroll
  for (int i = 0; i < 64; ++i) hv[i] = h2[(b * Hd + i) * Nd + n];
  float acc = b2[0];
  for (int k = 0; k < 64; ++k) {
    float z = b1[k];
#pragma unroll
    for (int i = 0; i < 64; ++i) z += hv[i] * W1[i * Hd + k];
    acc += gelu_f(z) * W2[k];
  }
  out[g] = acc;
}

// ---------------------------------------------------------------------------
extern "C" void kernel_launch(void* const* d_in, const int* in_sizes, int n_in,
                              void* d_out, int out_size, void* d_ws, size_t ws_size,
                              hipStream_t stream) {
  (void)in_sizes; (void)n_in; (void)out_size; (void)ws_size;
  const float* x   = (const float*)d_in[0];
  const float* W0  = (const float*)d_in[1];
  const float* b0  = (const float*)d_in[2];
  const float* wp  = (const float*)d_in[3];
  const float* wpi = (const float*)d_in[4];
  const float* wr  = (const float*)d_in[5];
  const float* wri = (const float*)d_in[6];
  const float* Wl  = (const float*)d_in[7];
  const float* bl  = (const float*)d_in[8];
  const float* W1  = (const float*)d_in[9];
  const float* b1  = (const float*)d_in[10];
  const float* W2  = (const float*)d_in[11];
  const float* b2  = (const float*)d_in[12];
  float* out = (float*)d_out;

  float* W = (float*)d_ws;
  size_t off = 0;
  float* h     = W + off; off += 524288;   // (B,H,N)
  float* are   = W + off; off += 524288;   // Re(alpha)
  float* aim   = W + off; off += 524288;   // Im(alpha)
  float* pre_t = W + off; off += 65536;    // poles/res transposed (p,i,o)
  float* pim_t = W + off; off += 65536;
  float* rre_t = W + off; off += 65536;
  float* rim_t = W + off; off += 65536;
  float* r1re  = W + off; off += 524288;   // out_res1
  float* r1im  = W + off; off += 524288;
  float* a2re  = W + off; off += 8192;     // out_res2 accumulator (B,H,P)
  float* a2im  = W + off; off += 8192;
  float* x1re  = W + off; off += 524288;   // Re(ifft(out_res1))
  float* x1im  = W + off; off += 524288;   // discarded
  float* Ere   = W + off; off += 1048576;  // (H,P,N)
  float* Eim   = W + off; off += 1048576;
  float* h2    = W + off; off += 524288;

  prep_params_kernel<<<256, 256, 0, stream>>>(wp, wpi, wr, wri,
                                              pre_t, pim_t, rre_t, rim_t,
                                              a2re, a2im);
  lift_kernel<<<2048, 256, 0, stream>>>(x, W0, b0, h);
  fft1024_kernel<<<512, 256, 0, stream>>>(h, nullptr, are, aim, -1.0f, 1.0f);
  laplace_freq_kernel<<<256, 128, 0, stream>>>(are, aim,
                                               pre_t, pim_t, rre_t, rim_t,
                                               r1re, r1im, a2re, a2im);
  fft1024_kernel<<<512, 256, 0, stream>>>(r1re, r1im, x1re, x1im,
                                          1.0f, 1.0f / 1024.0f);
  eterm_kernel<<<4096, 256, 0, stream>>>(pre_t, pim_t, Ere, Eim);
  combine_kernel<<<2048, 256, 0, stream>>>(x1re, a2re, a2im, Ere, Eim,
                                           h, Wl, bl, h2);
  project_kernel<<<32, 256, 0, stream>>>(h2, W1, b1, W2, b2, out);
}